// ODEFunc_52390011077014
// MI455X (gfx1250) — compile-verified
//
#include <hip/hip_runtime.h>
#include <hip/hip_bf16.h>

#define Tn 1024
#define Bn 256
#define TILE 16
#define WAVES 8
#define WGSIZE (WAVES * 32)

typedef float v2f __attribute__((ext_vector_type(2)));
typedef float v8f __attribute__((ext_vector_type(8)));

// ---------------------------------------------------------------------------
// Kernel 1: tiny MLP producing the memory kernel me[t] = sigmoid(MLP(t)).
// ---------------------------------------------------------------------------
__global__ void sir_mlp_kernel(const float* __restrict__ t,
                               const float* __restrict__ w1, const float* __restrict__ b1,
                               const float* __restrict__ w2, const float* __restrict__ b2,
                               const float* __restrict__ w3, const float* __restrict__ b3,
                               const float* __restrict__ w4, const float* __restrict__ b4,
                               float* __restrict__ me) {
    int i = blockIdx.x * blockDim.x + threadIdx.x;
    if (i >= Tn) return;
    float tv = t[i];
    float h[20], g[20];
    #pragma unroll
    for (int k = 0; k < 20; ++k) h[k] = tanhf(tv * w1[k] + b1[k]);
    for (int k = 0; k < 20; ++k) {
        float s = b2[k];
        #pragma unroll
        for (int m = 0; m < 20; ++m) s += h[m] * w2[m * 20 + k];
        g[k] = tanhf(s);
    }
    for (int k = 0; k < 20; ++k) {
        float s = b3[k];
        #pragma unroll
        for (int m = 0; m < 20; ++m) s += g[m] * w3[m * 20 + k];
        h[k] = tanhf(s);
    }
    float s = b4[0];
    #pragma unroll
    for (int k = 0; k < 20; ++k) s += h[k] * w4[k];
    me[i] = 1.0f / (1.0f + expf(-s));
}

// ---------------------------------------------------------------------------
// Kernel 2: blocked scan. 8-wave workgroup per 16-column batch tile.
//  - history GEMM split over 8 waves (disjoint K ranges), each wave runs a
//    2-accumulator unrolled V_WMMA_F32_16X16X4_F32 loop
//  - cooperative 256-thread reduction of the 8 partial tiles (dt-scaled)
//  - wave-0 lanes 0..15 run the 16 serial steps with register-resident
//    in-block history and me-tail
// ---------------------------------------------------------------------------
__global__ void __launch_bounds__(WGSIZE) sir_scan_kernel(
        const float* __restrict__ y,       // (Bn,1,3)
        const float* __restrict__ t,       // (Tn,1)
        const float* __restrict__ beta_p,
        const float* __restrict__ gamma_p,
        const float* __restrict__ me,      // (Tn)
        float* __restrict__ Ih,            // (Tn, Bn) workspace
        float* __restrict__ sol,           // (Tn, Bn, 3)
        float* __restrict__ dif)           // (Tn, Bn, 3)
{
    __shared__ float ldsPart[WAVES][TILE * TILE]; // per-wave partial tiles
    __shared__ float ldsP[TILE * TILE];           // reduced, dt-scaled tile

    const int tid  = threadIdx.x;          // 0..255
    const int lane = tid & 31;
    const int wv   = tid >> 5;             // wave id 0..7
    const int n    = lane & 15;            // tile column / A row / C column
    const int half = lane >> 4;            // 0 or 1
    const int col  = blockIdx.x * TILE + n;

    const float dt    = t[0] - t[1];
    const float beta  = beta_p[0];
    const float gamma = gamma_p[0];

    float S  = y[col * 3 + 0];
    float Iv = y[col * 3 + 1];
    float R  = y[col * 3 + 2];

    // me tail for the in-block triangular correction (me[Tn-16+x])
    float me_tail[16];
    #pragma unroll
    for (int x = 0; x < 16; ++x) me_tail[x] = me[Tn - 16 + x];

    for (int q = 0; q < Tn / TILE; ++q) {
        const int J = q * TILE;

        // ---- history GEMM, K split across 8 waves in 4-wide chunks
        const int C  = J >> 2;                       // total K chunks
        const int c0 = (wv * C) / WAVES;
        const int c1 = ((wv + 1) * C) / WAVES;
        const int aBase = Tn - J - n + 2 * half;     // A: M=lane%16, K=2*half+v
        const int kOff  = 2 * half;                  // B: N=lane%16, K=2*half+v

        v8f acc0 = {0.f, 0.f, 0.f, 0.f, 0.f, 0.f, 0.f, 0.f};
        v8f acc1 = {0.f, 0.f, 0.f, 0.f, 0.f, 0.f, 0.f, 0.f};

        int c = c0;
        for (; c + 2 <= c1; c += 2) {
            const int ta = 4 * c;
            const int tb = 4 * c + 4;
            v2f aA, bA, aB, bB;
            aA.x = me[aBase + ta];
            aA.y = me[aBase + ta + 1];
            bA.x = Ih[(ta + kOff) * Bn + col];
            bA.y = Ih[(ta + kOff + 1) * Bn + col];
            aB.x = me[aBase + tb];
            aB.y = me[aBase + tb + 1];
            bB.x = Ih[(tb + kOff) * Bn + col];
            bB.y = Ih[(tb + kOff + 1) * Bn + col];
            acc0 = __builtin_amdgcn_wmma_f32_16x16x4_f32(
                    false, aA, false, bA, (short)0, acc0, false, false);
            acc1 = __builtin_amdgcn_wmma_f32_16x16x4_f32(
                    false, aB, false, bB, (short)0, acc1, false, false);
        }
        if (c < c1) {
            const int ta = 4 * c;
            v2f aA, bA;
            aA.x = me[aBase + ta];
            aA.y = me[aBase + ta + 1];
            bA.x = Ih[(ta + kOff) * Bn + col];
            bA.y = Ih[(ta + kOff + 1) * Bn + col];
            acc0 = __builtin_amdgcn_wmma_f32_16x16x4_f32(
                    false, aA, false, bA, (short)0, acc0, false, false);
        }
        acc0 = acc0 + acc1;

        // C/D layout: vgpr v -> row M = v + 8*half, column N = n
        #pragma unroll
        for (int v = 0; v < 8; ++v) {
            ldsPart[wv][(v + 8 * half) * TILE + n] = acc0[v];
        }
        __syncthreads();

        // ---- cooperative reduction of 8 partial tiles, dt-scaled
        {
            float p = 0.f;
            #pragma unroll
            for (int w = 0; w < WAVES; ++w) p += ldsPart[w][tid];
            ldsP[tid] = p * dt;
        }
        __syncthreads();

        // ---- serial 16 steps (wave 0, lanes 0..15; one batch column each)
        if (tid < 16) {
            float Iblk[16];
            #pragma unroll
            for (int r = 0; r < 16; ++r) {
                const int j = J + r;
                if (j == 0) {
                    sol[col * 3 + 0] = S;
                    sol[col * 3 + 1] = Iv;
                    sol[col * 3 + 2] = R;
                    Ih[col] = Iv;
                    Iblk[0] = Iv;
                    continue;
                }
                float integ = ldsP[r * TILE + n];
                #pragma unroll
                for (int u = 0; u < r; ++u) {
                    integ += dt * me_tail[16 - r + u] * Iblk[u];
                }
                const float bSI = beta * S * Iv;
                const float gI  = gamma * Iv;
                const float d0 = -bSI + integ;
                const float d1 =  bSI - gI;
                const float d2 =  gI - integ;
                S  += d0 * dt;
                Iv += d1 * dt;
                R  += d2 * dt;
                sol[j * Bn * 3 + col * 3 + 0] = S;
                sol[j * Bn * 3 + col * 3 + 1] = Iv;
                sol[j * Bn * 3 + col * 3 + 2] = R;
                dif[(j - 1) * Bn * 3 + col * 3 + 0] = d0;
                dif[(j - 1) * Bn * 3 + col * 3 + 1] = d1;
                dif[(j - 1) * Bn * 3 + col * 3 + 2] = d2;
                Ih[j * Bn + col] = Iv;          // history for later blocks
                Iblk[r] = Iv;                   // in-block history (registers)
            }
            __threadfence_block();              // publish Ih rows to all waves
        }
        __syncthreads();
    }

    // diff[T-1] = 0
    if (tid < 16) {
        dif[(Tn - 1) * Bn * 3 + col * 3 + 0] = 0.f;
        dif[(Tn - 1) * Bn * 3 + col * 3 + 1] = 0.f;
        dif[(Tn - 1) * Bn * 3 + col * 3 + 2] = 0.f;
    }
}

extern "C" void kernel_launch(void* const* d_in, const int* in_sizes, int n_in,
                              void* d_out, int out_size, void* d_ws, size_t ws_size,
                              hipStream_t stream) {
    const float* t_arr = (const float*)d_in[0];
    const float* y     = (const float*)d_in[1];
    const float* w1    = (const float*)d_in[2];
    const float* b1    = (const float*)d_in[3];
    const float* w2    = (const float*)d_in[4];
    const float* b2    = (const float*)d_in[5];
    const float* w3    = (const float*)d_in[6];
    const float* b3    = (const float*)d_in[7];
    const float* w4    = (const float*)d_in[8];
    const float* b4    = (const float*)d_in[9];
    const float* beta  = (const float*)d_in[10];
    const float* gamma = (const float*)d_in[11];

    float* me  = (float*)d_ws;            // Tn floats
    float* Ih  = (float*)d_ws + Tn;       // Tn*Bn floats
    float* sol = (float*)d_out;           // Tn*Bn*3
    float* dif = (float*)d_out + Tn * Bn * 3;

    sir_mlp_kernel<<<Tn / 256, 256, 0, stream>>>(t_arr, w1, b1, w2, b2, w3, b3, w4, b4, me);
    sir_scan_kernel<<<Bn / TILE, WGSIZE, 0, stream>>>(y, t_arr, beta, gamma, me, Ih, sol, dif);
}